// GPTNeoXRoutedMLP_29600914604179
// MI455X (gfx1250) — compile-verified
//
#include <hip/hip_runtime.h>
#include <hip/hip_bf16.h>
#include <math.h>

// ---------------------------------------------------------------------------
// MoE routed MLP for MI455X (gfx1250, wave32, WMMA).
//   H=1024, F=4096, E=8, top-2, N = 2048 tokens, all fp32.
// Memory-bound (~268 MB of weights @ 23.3 TB/s ~= 11.5us HBM floor):
//   - sparse routing (only top-2 experts per token -> 4x less compute),
//   - fused GEMM1 -> GELU -> GEMM2 (16..32 x F intermediate stays in LDS),
//   - fp32 WMMA (v_wmma_f32_16x16x4_f32) keeps reference fp32 precision,
//   - M=32 token tiles: halves per-expert weight re-reads from L2 and gives
//     2x register reuse of every B fragment (two M-subtiles per B load),
//   - LDS row padding (+16B) kills the 16-way bank conflict on A-fragment
//     ds_load_2addr_b32 (row strides were multiples of the 256B bank period).
// ---------------------------------------------------------------------------

#define HDIM 1024
#define FDIM 4096
#define NEXP 8
#define NTOK 2048
#define FCHK 128           // F columns per outer iteration (8 waves x 16)
#define MT   32            // tokens per workgroup tile (2 x 16-row subtiles)
#define XSTR (HDIM + 4)    // padded xs row stride (4112 B: 16B-aligned, no bank clash)
#define HSTR (FCHK + 4)    // padded hs row stride (528 B)

typedef __attribute__((ext_vector_type(2))) float v2f;
typedef __attribute__((ext_vector_type(8))) float v8f;

__device__ __forceinline__ v8f wmma_f32(v2f a, v2f b, v8f c) {
  // 8-arg form: (neg_a, A, neg_b, B, c_mod, C, reuse_a, reuse_b)
  return __builtin_amdgcn_wmma_f32_16x16x4_f32(false, a, false, b, (short)0, c,
                                               false, false);
}

// ---------------------------------------------------------------------------
// Kernel 1: router. One wave32 per token: logits = x[n] @ router_w, top-2,
// softmax over the two logits, append (token, score) into per-expert buckets.
// ---------------------------------------------------------------------------
__global__ __launch_bounds__(256) void router_kernel(
    const float* __restrict__ x, const float* __restrict__ rw,
    int* __restrict__ cnt, int* __restrict__ btok, float* __restrict__ bsc) {
  const int wave = threadIdx.x >> 5;
  const int lane = threadIdx.x & 31;
  const int n = blockIdx.x * 8 + wave;
  if (n >= NTOK) return;

  float acc[NEXP];
#pragma unroll
  for (int e = 0; e < NEXP; ++e) acc[e] = 0.0f;

  const float* xr = x + (size_t)n * HDIM;
  for (int h = lane; h < HDIM; h += 32) {
    float xv = xr[h];
    const float* wr = rw + (size_t)h * NEXP;
#pragma unroll
    for (int e = 0; e < NEXP; ++e) acc[e] = fmaf(xv, wr[e], acc[e]);
  }
#pragma unroll
  for (int e = 0; e < NEXP; ++e) {
#pragma unroll
    for (int off = 16; off > 0; off >>= 1)
      acc[e] += __shfl_xor(acc[e], off, 32);
  }

  if (lane == 0) {
    // top-2, lowest index wins ties (matches jax.lax.top_k)
    int i0 = 0;
    float l0 = acc[0];
#pragma unroll
    for (int e = 1; e < NEXP; ++e)
      if (acc[e] > l0) { l0 = acc[e]; i0 = e; }
    int i1 = (i0 == 0) ? 1 : 0;
    float l1 = acc[i1];
#pragma unroll
    for (int e = 0; e < NEXP; ++e)
      if (e != i0 && acc[e] > l1) { l1 = acc[e]; i1 = e; }

    // softmax over {l0, l1}, l0 >= l1
    float e1 = __expf(l1 - l0);
    float inv = 1.0f / (1.0f + e1);

    int p0 = atomicAdd(&cnt[i0], 1);
    btok[i0 * NTOK + p0] = n;
    bsc[i0 * NTOK + p0] = inv;
    int p1 = atomicAdd(&cnt[i1], 1);
    btok[i1 * NTOK + p1] = n;
    bsc[i1 * NTOK + p1] = e1 * inv;
  }
}

// ---------------------------------------------------------------------------
// Kernel 2: fused expert MLP on a 32-token tile of one expert's bucket.
// 256 threads = 8 wave32s. Wave w owns output columns [w*128, w*128+128)
// for both 16-row M-subtiles: 16 persistent 16x16 f32 WMMA accumulators.
// ---------------------------------------------------------------------------
__global__ __launch_bounds__(256) void moe_mlp_kernel(
    const float* __restrict__ x, const float* __restrict__ w1,
    const float* __restrict__ b1, const float* __restrict__ w2,
    const float* __restrict__ b2, const int* __restrict__ cnt,
    const int* __restrict__ btok, const float* __restrict__ bsc,
    float* __restrict__ out) {
  const int e = blockIdx.y;
  const int base = blockIdx.x * MT;
  const int cn = cnt[e];
  if (base >= cn) return;  // uniform early-exit; EXEC stays full past here

  const int tid = threadIdx.x;
  const int wave = tid >> 5;
  const int lane = tid & 31;
  const int lh = lane & 15;           // N (col) index within a 16-wide tile
  const int khi = (lane >> 4) << 1;   // A/B K offset: 0 (lanes 0-15) / 2 (16-31)
  const int mhi = (lane >> 4) << 3;   // C row offset: 0 / 8

  __shared__ int s_tok[MT];
  __shared__ float s_sc[MT];
  extern __shared__ float smem[];     // xs[MT*XSTR] + hs[MT*HSTR] (~145 KB)
  float* xs = smem;
  float* hs = smem + MT * XSTR;

  if (tid < MT) {
    int idx = base + tid;
    if (idx < cn) {
      s_tok[tid] = btok[e * NTOK + idx];
      s_sc[tid] = bsc[e * NTOK + idx];
    } else {
      s_tok[tid] = btok[e * NTOK];  // valid pad row; masked in epilogue
      s_sc[tid] = 0.0f;
    }
  }
  __syncthreads();

  // Cooperative X-tile load: 32 rows x 1024 f32 (row stride padded by 16 B)
  for (int i = tid; i < MT * (HDIM / 4); i += 256) {
    int row = i >> 8;   // 256 float4 per row
    int c4 = i & 255;
    ((float4*)(xs + row * XSTR))[c4] =
        ((const float4*)(x + (size_t)s_tok[row] * HDIM))[c4];
  }

  const float* w1e = w1 + (size_t)e * HDIM * FDIM;
  const float* w2e = w2 + (size_t)e * FDIM * HDIM;

  v8f acc[2][8];
#pragma unroll
  for (int m = 0; m < 2; ++m)
#pragma unroll
    for (int j = 0; j < 8; ++j) acc[m][j] = (v8f){0, 0, 0, 0, 0, 0, 0, 0};

  const float* xr0 = xs + lh * XSTR;         // A rows, subtile 0 (M=0..15)
  const float* xr1 = xs + (16 + lh) * XSTR;  // A rows, subtile 1 (M=16..31)
  const float* hr0 = hs + lh * HSTR;
  const float* hr1 = hs + (16 + lh) * HSTR;

  for (int fb = 0; fb < FDIM; fb += FCHK) {
    __syncthreads();  // hs from previous chunk fully consumed

    // ---- GEMM1: [32x16] = X[32x1024] @ w1[:, fc0:fc0+16]; each B fragment
    //      feeds two WMMAs (both M-subtiles).
    const int fc0 = fb + wave * 16;
    v8f c1a = (v8f){0, 0, 0, 0, 0, 0, 0, 0};
    v8f c1b = (v8f){0, 0, 0, 0, 0, 0, 0, 0};
    const float* wp1 = w1e + (size_t)khi * FDIM + fc0 + lh;
    for (int k = 0; k < HDIM; k += 4) {
      v2f b;
      b.x = wp1[(size_t)k * FDIM];
      b.y = wp1[(size_t)(k + 1) * FDIM];
      v2f a0, a1;
      a0.x = xr0[k + khi];
      a0.y = xr0[k + khi + 1];
      a1.x = xr1[k + khi];
      a1.y = xr1[k + khi + 1];
      c1a = wmma_f32(a0, b, c1a);
      c1b = wmma_f32(a1, b, c1b);
    }
    // prefetch next chunk of w1 (global_prefetch_b8)
    if (fb + FCHK < FDIM)
      __builtin_prefetch(w1e + (size_t)khi * FDIM + fc0 + FCHK + lh, 0, 0);

    // ---- bias + exact GELU, stage h tile (32 x 128, padded) to LDS
    {
      const int col = fc0 + lh;
      const float bias = b1[e * FDIM + col];
      const int hc = wave * 16 + lh;
#pragma unroll
      for (int r = 0; r < 8; ++r) {
        float v0 = c1a[r] + bias;
        float v1 = c1b[r] + bias;
        v0 = 0.5f * v0 * (1.0f + erff(v0 * 0.70710678118654752f));
        v1 = 0.5f * v1 * (1.0f + erff(v1 * 0.70710678118654752f));
        hs[(mhi + r) * HSTR + hc] = v0;
        hs[(16 + mhi + r) * HSTR + hc] = v1;
      }
    }
    __syncthreads();

    // ---- GEMM2 partial: acc[32x128] += h[32x128] @ w2[fb:fb+128, w*128:+128]
    for (int k = 0; k < FCHK; k += 4) {
      v2f a0, a1;
      a0.x = hr0[k + khi];
      a0.y = hr0[k + khi + 1];
      a1.x = hr1[k + khi];
      a1.y = hr1[k + khi + 1];
      const int fr = fb + k + khi;
      const float* wp2 = w2e + (size_t)fr * HDIM + wave * 128 + lh;
#pragma unroll
      for (int j = 0; j < 8; ++j) {
        v2f b;
        b.x = wp2[j * 16];
        b.y = wp2[j * 16 + HDIM];
        acc[0][j] = wmma_f32(a0, b, acc[0][j]);
        acc[1][j] = wmma_f32(a1, b, acc[1][j]);
      }
    }
  }

  // ---- epilogue: + b2, scale by routing score, scatter-add into out rows.
  // Each token has exactly 2 expert contributions; float add of 2 terms is
  // commutative, so atomic order does not change the result.
#pragma unroll
  for (int j = 0; j < 8; ++j) {
    const int col = wave * 128 + j * 16 + lh;
    const float bias = b2[e * HDIM + col];
#pragma unroll
    for (int m = 0; m < 2; ++m) {
#pragma unroll
      for (int r = 0; r < 8; ++r) {
        int mrow = m * 16 + mhi + r;
        if (base + mrow < cn) {
          float v = (acc[m][j][r] + bias) * s_sc[mrow];
          atomicAdd(out + (size_t)s_tok[mrow] * HDIM + col, v);
        }
      }
    }
  }
}

// ---------------------------------------------------------------------------
extern "C" void kernel_launch(void* const* d_in, const int* in_sizes, int n_in,
                              void* d_out, int out_size, void* d_ws,
                              size_t ws_size, hipStream_t stream) {
  const float* hidden = (const float*)d_in[0];  // [S,B,H] = [2048, H] flat
  const float* rw     = (const float*)d_in[1];  // [H,E]
  const float* w1     = (const float*)d_in[2];  // [E,H,F]
  const float* b1     = (const float*)d_in[3];  // [E,F]
  const float* w2     = (const float*)d_in[4];  // [E,F,H]
  const float* b2     = (const float*)d_in[5];  // [E,H]
  float* out = (float*)d_out;

  // workspace: cnt[8] | btok[E*N] | bsc[E*N]
  int* cnt = (int*)d_ws;
  int* btok = (int*)((char*)d_ws + 256);
  float* bsc = (float*)((char*)d_ws + 256 + (size_t)NEXP * NTOK * sizeof(int));

  hipMemsetAsync(d_out, 0, (size_t)out_size * sizeof(float), stream);
  hipMemsetAsync(cnt, 0, NEXP * sizeof(int), stream);

  router_kernel<<<NTOK / 8, 256, 0, stream>>>(hidden, rw, cnt, btok, bsc);

  dim3 grid(NTOK / MT, NEXP);
  size_t lds = (size_t)(MT * XSTR + MT * HSTR) * sizeof(float);  // ~145 KB
  moe_mlp_kernel<<<grid, 256, lds, stream>>>(hidden, w1, b1, w2, b2, cnt, btok,
                                             bsc, out);
}